// GHE_25555055411498
// MI455X (gfx1250) — compile-verified
//
#include <hip/hip_runtime.h>
#include <math.h>

// ---------------------------------------------------------------------------
// GMM log-likelihood loss pipeline for MI455X (gfx1250, wave32, WMMA).
// GEMMs use V_WMMA_F32_16X16X4_F32 (full fp32; workload is memory-bound:
// ~140MB @ 23.3TB/s ~ 6us, so the fp32 matrix path costs nothing and keeps
// parity with the fp32 reference). Inner loops are branch-free with b64
// vector loads and global_prefetch_b8 on the streaming operand.
// ---------------------------------------------------------------------------

typedef __attribute__((ext_vector_type(2))) float v2f;
typedef __attribute__((ext_vector_type(8))) float v8f;

#define BB     8
#define CC     64
#define NN     12288          // m*n*t
#define KK     16
#define FANIN  (CC * NN)      // 786432
#define LOG2PI 1.8378770664093453f

// ---------------------------------------------------------------- utilities
__device__ __forceinline__ float blockReduceSum256(float v, float* sh) {
  const int t = threadIdx.x;
  sh[t] = v;
  __syncthreads();
  for (int s = 128; s > 0; s >>= 1) {
    if (t < s) sh[t] += sh[t + s];
    __syncthreads();
  }
  return sh[0];
}

// ------------------------------------------------------------- zero scratch
__global__ void k_zero(float* __restrict__ z, int n) {
  const int i = blockIdx.x * blockDim.x + threadIdx.x;
  if (i < n) z[i] = 0.0f;
}

// ------------------------------------------------- per-point inverse norms
__global__ void k_norms(const float* __restrict__ rep,
                        const float* __restrict__ aug,
                        float* __restrict__ invr, float* __restrict__ inva) {
  const int idx = blockIdx.x * blockDim.x + threadIdx.x;
  if (idx >= BB * NN) return;
  const int b = idx / NN;
  const int p = idx - b * NN;
  const float* r0 = rep + (size_t)b * FANIN + p;
  const float* a0 = aug + (size_t)b * FANIN + p;
  float sr = 0.0f, sa = 0.0f;
  for (int c = 0; c < CC; ++c) {
    float x = r0[c * NN]; sr += x * x;
    float y = a0[c * NN]; sa += y * y;
  }
  invr[idx] = 1.0f / fmaxf(sqrtf(sr), 1e-12f);
  inva[idx] = 1.0f / fmaxf(sqrtf(sa), 1e-12f);
}

// ------------------------------------------------------------ alpha logits
// logits[b,k] = sum_K aug_norm[b,K] * W_alpha[k,K].  One 16x16 f32 WMMA tile
// (rows 8..15 masked to zero), split-K over 1024 waves, 768 K-elems per wave
// (768 | NN, so a chunk never crosses a channel slab: c/p hoisted out).
#define ALPHA_BLOCKS 128
#define ALPHA_TPB    256
#define ALPHA_CHUNK  768
__global__ void k_alpha_gemm(const float* __restrict__ aug,
                             const float* __restrict__ inva,
                             const float* __restrict__ Wa,
                             float* __restrict__ logits) {
  const int lane = threadIdx.x & 31;
  const int wave = (blockIdx.x * blockDim.x + threadIdx.x) >> 5;
  const int hi   = lane >> 4;       // ISA A 16x4 layout: K={0,1} / K={2,3}
  const int low  = lane & 15;
  const int kstart = wave * ALPHA_CHUNK;
  const int cslab  = kstart / NN;                 // loop-invariant
  const int pbase  = kstart - cslab * NN;
  const float maskf = (low < BB) ? 1.0f : 0.0f;   // pad rows 8..15 -> 0
  const int   row   = low & 7;                    // always-in-bounds source

  const float* aPtr = aug + (size_t)row * FANIN + kstart + hi * 2;
  const float* iPtr = inva + row * NN + pbase + hi * 2;
  const float* bPtr = Wa + (size_t)low * FANIN + kstart + hi * 2;

  v8f acc = {};
  for (int s = 0; s < ALPHA_CHUNK / 4; ++s) {     // 192 WMMA steps
    __builtin_prefetch(bPtr + s * 4 + 512, 0, 1); // global_prefetch_b8
    const v2f a2 = *(const v2f*)(aPtr + s * 4);
    const v2f i2 = *(const v2f*)(iPtr + s * 4);
    const v2f b2 = *(const v2f*)(bPtr + s * 4);
    const v2f a  = a2 * i2 * maskf;               // branch-free masking
    acc = __builtin_amdgcn_wmma_f32_16x16x4_f32(false, a, false, b2,
                                                (short)0, acc, false, false);
  }
  // C layout: VGPR j -> M = j + 8*hi, column = low
#pragma unroll
  for (int j = 0; j < 8; ++j) {
    const int M = j + hi * 8;
    if (M < BB) atomicAdd(&logits[M * KK + low], acc[j]);
  }
}

// ----------------------------------------------------- log-softmax of alpha
__global__ void k_softmax(const float* __restrict__ logits,
                          float* __restrict__ logalpha) {
  const int t = threadIdx.x;
  if (t < BB) {
    float mx = -1e30f;
    for (int k = 0; k < KK; ++k) mx = fmaxf(mx, logits[t * KK + k]);
    float s = 0.0f;
    for (int k = 0; k < KK; ++k) s += expf(logits[t * KK + k] - mx);
    const float ls = logf(s);
    for (int k = 0; k < KK; ++k)
      logalpha[t * KK + k] = logits[t * KK + k] - mx - ls;
  }
}

// ------------------------------------------------------------- mus / sigmas
// C[k,c] = sum_n W[k,n] * aug_norm[b,c,n].  Split-K (4 chunks of 3072) x
// {mu,sigma} x 4 c-tiles x 8 batches = 256 single-wave blocks; f32 atomic
// merge into zeroed accumulator, epilogue applied by k_musig_fin.
__global__ void k_musig_gemm(const float* __restrict__ aug,
                             const float* __restrict__ inva,
                             const float* __restrict__ Wmu,
                             const float* __restrict__ Wsig,
                             float* __restrict__ msacc) {
  const int bid   = blockIdx.x;
  const int kc    = bid & 3;          // K chunk
  const int which = (bid >> 2) & 1;   // 0: mu, 1: sigma
  const int ct    = (bid >> 3) & 3;   // c tile
  const int b     = bid >> 5;         // batch
  const int lane  = threadIdx.x & 31;
  const int hi    = lane >> 4;
  const int low   = lane & 15;

  const float* W  = which ? Wsig : Wmu;
  const int    c  = ct * 16 + low;
  const int    n0 = kc * (NN / 4);    // 3072-element K chunk

  const float* Arow = W + (size_t)low * NN + n0 + hi * 2;          // W[k,n]
  const float* Brow = aug + ((size_t)b * CC + c) * NN + n0 + hi * 2;
  const float* Inv  = inva + b * NN + n0 + hi * 2;

  v8f acc = {};
  for (int s = 0; s < NN / 16; ++s) {               // 768 WMMA steps
    __builtin_prefetch(Brow + s * 4 + 512, 0, 1);   // global_prefetch_b8
    const v2f a2 = *(const v2f*)(Arow + s * 4);
    const v2f b2 = *(const v2f*)(Brow + s * 4);
    const v2f i2 = *(const v2f*)(Inv + s * 4);
    const v2f bn = b2 * i2;                         // normalize on the fly
    acc = __builtin_amdgcn_wmma_f32_16x16x4_f32(false, a2, false, bn,
                                                (short)0, acc, false, false);
  }
#pragma unroll
  for (int j = 0; j < 8; ++j) {
    const int kidx = j + hi * 8;
    atomicAdd(&msacc[(((size_t)which * BB + b) * KK + kidx) * CC + c], acc[j]);
  }
}

// epilogue: bias + exp folding; stores mu, 1/sigma, -log(sigma)-0.5*log(2pi)
__global__ void k_musig_fin(const float* __restrict__ msacc,
                            const float* __restrict__ bmu,
                            const float* __restrict__ bsig,
                            float* __restrict__ mu,
                            float* __restrict__ lcoef,
                            float* __restrict__ ivs) {
  const int idx = blockIdx.x * blockDim.x + threadIdx.x;
  if (idx >= BB * KK * CC) return;
  const int k = (idx >> 6) & 15;
  const float zmu = msacc[idx] + bmu[k];
  const float zsg = msacc[BB * KK * CC + idx] + bsig[k];
  mu[idx]    = zmu;
  ivs[idx]   = expf(-zsg);             // 1/sigma   (sigma = exp(z))
  lcoef[idx] = -zsg - 0.5f * LOG2PI;   // -log(sigma) - 0.5*log(2pi)
}

// ------------------------------------------ GMM log-pdf + product over c
__global__ void k_lcp(const float* __restrict__ rep,
                      const float* __restrict__ invr,
                      const float* __restrict__ mu,
                      const float* __restrict__ lcoef,
                      const float* __restrict__ ivs,
                      float* __restrict__ P,
                      float* __restrict__ sumsq) {
  const int b  = blockIdx.y;
  const int p0 = blockIdx.x * 16;
  __shared__ float sh_h[CC][16];
  __shared__ float sh_mu[KK][CC + 1];
  __shared__ float sh_lc[KK][CC + 1];
  __shared__ float sh_iv[KK][CC + 1];
  __shared__ float red[16][17];

  const int t = threadIdx.x;
  for (int i = t; i < CC * 16; i += 256) {
    const int c = i >> 4, pl = i & 15;
    sh_h[c][pl] = rep[((size_t)b * CC + c) * NN + p0 + pl] *
                  invr[b * NN + p0 + pl];
  }
  for (int i = t; i < KK * CC; i += 256) {
    const int k = i >> 6, c = i & 63;
    const size_t o = (size_t)b * KK * CC + i;
    sh_mu[k][c] = mu[o];
    sh_lc[k][c] = lcoef[o];
    sh_iv[k][c] = ivs[o];
  }
  __syncthreads();

  const int k = t & 15, pl = t >> 4;
  float prod = 1.0f;
#pragma unroll 8
  for (int c = 0; c < CC; ++c) {
    const float d = (sh_h[c][pl] - sh_mu[k][c]) * sh_iv[k][c];
    prod *= (sh_lc[k][c] - 0.5f * d * d);
  }
  P[((size_t)b * NN + p0 + pl) * KK + k] = prod;

  red[pl][k] = prod * prod;
  __syncthreads();
  if (pl == 0) {
    float s = 0.0f;
#pragma unroll
    for (int i = 0; i < 16; ++i) s += red[i][k];
    atomicAdd(&sumsq[b * KK + k], s);
  }
}

// ------------------------------------------------------------- final loss
__global__ void k_loss(const float* __restrict__ P,
                       const float* __restrict__ sumsq,
                       const float* __restrict__ logalpha,
                       float* __restrict__ lossacc) {
  __shared__ float sh[256];
  const int idx = blockIdx.x * blockDim.x + threadIdx.x;
  float val = 0.0f;
  if (idx < BB * NN) {
    const int b = idx / NN;
    float s = 0.0f;
#pragma unroll
    for (int k = 0; k < KK; ++k) {
      const float inv = 1.0f / fmaxf(sqrtf(sumsq[b * KK + k]), 1e-12f);
      const float lp  = P[(size_t)idx * KK + k] * inv + logalpha[b * KK + k];
      s += expf(lp);
    }
    val = logf(s);
  }
  const float tot = blockReduceSum256(val, sh);
  if (threadIdx.x == 0) atomicAdd(lossacc, tot);
}

__global__ void k_write(const float* __restrict__ lossacc,
                        float* __restrict__ out) {
  if (threadIdx.x == 0) out[0] = -lossacc[0] / (float)(BB * NN);
}

// ---------------------------------------------------------------- launcher
extern "C" void kernel_launch(void* const* d_in, const int* in_sizes, int n_in,
                              void* d_out, int out_size, void* d_ws,
                              size_t ws_size, hipStream_t stream) {
  (void)in_sizes; (void)n_in; (void)out_size; (void)ws_size;
  const float* rep  = (const float*)d_in[0];
  const float* aug  = (const float*)d_in[1];
  const float* Wa   = (const float*)d_in[2];
  const float* Wmu  = (const float*)d_in[3];
  const float* bmu  = (const float*)d_in[4];
  const float* Wsig = (const float*)d_in[5];
  const float* bsig = (const float*)d_in[6];
  float* out = (float*)d_out;

  // workspace carve-up (~7.3 MB, re-initialized every call; zeroed arrays
  // are contiguous so a single one-thread-per-element zero kernel covers them)
  float* ws       = (float*)d_ws;
  float* invr     = ws; ws += BB * NN;             // 98304
  float* inva     = ws; ws += BB * NN;             // 98304
  float* zbase    = ws;
  float* logits   = ws; ws += 256;                 // padded 16x16 tile
  float* logalpha = ws; ws += 128;
  float* sumsq    = ws; ws += 128;
  float* lossacc  = ws; ws += 8;                   // padded
  float* msacc    = ws; ws += 2 * BB * KK * CC;    // 16384
  const int zcount = (int)(ws - zbase);            // 16904
  float* mu       = ws; ws += BB * KK * CC;        // 8192
  float* lcoef    = ws; ws += BB * KK * CC;        // 8192
  float* ivs      = ws; ws += BB * KK * CC;        // 8192
  float* P        = ws;                            // 8*12288*16 = 1572864

  k_zero<<<(zcount + 255) / 256, 256, 0, stream>>>(zbase, zcount);
  k_norms<<<(BB * NN + 255) / 256, 256, 0, stream>>>(rep, aug, invr, inva);
  k_alpha_gemm<<<ALPHA_BLOCKS, ALPHA_TPB, 0, stream>>>(aug, inva, Wa, logits);
  k_softmax<<<1, 32, 0, stream>>>(logits, logalpha);
  k_musig_gemm<<<256, 32, 0, stream>>>(aug, inva, Wmu, Wsig, msacc);
  k_musig_fin<<<(BB * KK * CC + 255) / 256, 256, 0, stream>>>(msacc, bmu, bsig,
                                                              mu, lcoef, ivs);
  k_lcp<<<dim3(NN / 16, BB), 256, 0, stream>>>(rep, invr, mu, lcoef, ivs,
                                               P, sumsq);
  k_loss<<<(BB * NN + 255) / 256, 256, 0, stream>>>(P, sumsq, logalpha,
                                                    lossacc);
  k_write<<<1, 32, 0, stream>>>(lossacc, out);
}